// ScaledDotAttention_16192026706642
// MI455X (gfx1250) — compile-verified
//
#include <hip/hip_runtime.h>

typedef float v2f __attribute__((ext_vector_type(2)));
typedef float v8f __attribute__((ext_vector_type(8)));

#define SEQ       2048
#define HEADS     16
#define DIM       64
#define KTILES    (SEQ / 16)
#define ROWSTRIDE (HEADS * DIM)   // 1024 floats between consecutive seq positions

// One wave (32 lanes) handles one 16-row query tile of one (batch, head).
__global__ __launch_bounds__(32) void
attn_fwd_wmma(const float* __restrict__ qg, const float* __restrict__ kg,
              const float* __restrict__ vg, float* __restrict__ ctx_out,
              float* __restrict__ attn_out)
{
    __shared__ float pbuf[16 * 17];   // padded 16x16 P tile for C->A relayout

    const int wid = blockIdx.x;            // 0 .. 2*16*128-1
    const int qt  = wid & (KTILES - 1);    // query tile
    const int h   = (wid >> 7) & (HEADS - 1);
    const int n   = wid >> 11;             // batch

    const int lane = threadIdx.x;          // 0..31 (wave32)
    const int half = lane >> 4;            // 0 | 1
    const int l15  = lane & 15;

    const size_t batch_stride = (size_t)SEQ * ROWSTRIDE;
    const float* qb = qg + (size_t)n * batch_stride + (size_t)(qt * 16) * ROWSTRIDE + (size_t)h * DIM;
    const float* kb = kg + (size_t)n * batch_stride + (size_t)h * DIM;
    const float* vb = vg + (size_t)n * batch_stride + (size_t)h * DIM;
    float* ab = attn_out + (((size_t)(n * HEADS + h)) * SEQ + (size_t)(qt * 16)) * SEQ;
    float* cb = ctx_out  + (size_t)n * batch_stride + (size_t)(qt * 16) * ROWSTRIDE + (size_t)h * DIM;

    // ---- load Q tile as 16 A-fragments (16x4 f32 each) ----
    // A layout: M = lane&15 for both halves; lanes0-15 hold K={0,1}, lanes16-31 K={2,3}.
    v2f qa[16];
    {
        const float* qrow = qb + (size_t)l15 * ROWSTRIDE;
        #pragma unroll
        for (int f = 0; f < 16; ++f) {
            const float2 t = *(const float2*)(qrow + 4 * f + 2 * half);
            qa[f] = (v2f){t.x, t.y};
        }
    }

    const float NEG_INF = -__builtin_inff();
    float mrow[8], lrow[8];
    #pragma unroll
    for (int r = 0; r < 8; ++r) { mrow[r] = NEG_INF; lrow[r] = 0.0f; }

    // ================= pass 1: online softmax statistics =================
    for (int kt = 0; kt < KTILES; ++kt) {
        const float* krow = kb + (size_t)(kt * 16 + l15) * ROWSTRIDE;
        v8f c = {};
        #pragma unroll
        for (int f = 0; f < 16; ++f) {
            const float2 t = *(const float2*)(krow + 4 * f + 2 * half);
            v2f bfrag = (v2f){t.x, t.y};   // B: N = key = kt*16 + l15, K rows per half
            c = __builtin_amdgcn_wmma_f32_16x16x4_f32(false, qa[f], false, bfrag,
                                                      (short)0, c, false, false);
        }
        #pragma unroll
        for (int r = 0; r < 8; ++r) {      // C layout: row = r + 8*half, col = l15
            float s  = c[r] * 0.125f;      // 1/sqrt(64)
            float mo = mrow[r];
            float mn = fmaxf(mo, s);
            lrow[r]  = lrow[r] * __expf(mo - mn) + __expf(s - mn);
            mrow[r]  = mn;
        }
    }

    // merge (m, l) across the 16 column-lanes of each half-wave
    #pragma unroll
    for (int mask = 1; mask <= 8; mask <<= 1) {
        #pragma unroll
        for (int r = 0; r < 8; ++r) {
            float mo = __shfl_xor(mrow[r], mask, 32);
            float lo = __shfl_xor(lrow[r], mask, 32);
            float mn = fmaxf(mrow[r], mo);
            lrow[r]  = lrow[r] * __expf(mrow[r] - mn) + lo * __expf(mo - mn);
            mrow[r]  = mn;
        }
    }
    float linv[8];
    #pragma unroll
    for (int r = 0; r < 8; ++r) linv[r] = 1.0f / lrow[r];

    // ================= pass 2: normalize, emit attn, accumulate P@V =================
    v8f ctxacc[4];
    #pragma unroll
    for (int dt = 0; dt < 4; ++dt) ctxacc[dt] = (v8f){};

    for (int kt = 0; kt < KTILES; ++kt) {
        // recompute score tile
        const float* krow = kb + (size_t)(kt * 16 + l15) * ROWSTRIDE;
        v8f c = {};
        #pragma unroll
        for (int f = 0; f < 16; ++f) {
            const float2 t = *(const float2*)(krow + 4 * f + 2 * half);
            v2f bfrag = (v2f){t.x, t.y};
            c = __builtin_amdgcn_wmma_f32_16x16x4_f32(false, qa[f], false, bfrag,
                                                      (short)0, c, false, false);
        }

        // normalize -> P, stream to attn (non-temporal), stage in LDS for relayout
        #pragma unroll
        for (int r = 0; r < 8; ++r) {
            float pr = __expf(c[r] * 0.125f - mrow[r]) * linv[r];
            __builtin_nontemporal_store(pr, ab + (size_t)(r + 8 * half) * SEQ
                                              + (size_t)(kt * 16 + l15));
            pbuf[(r + 8 * half) * 17 + l15] = pr;
        }
        __syncthreads();

        // ctx += P(16x16) @ V(16x64): 4 key-chunks x 4 d-tiles of 16x16x4 WMMA
        #pragma unroll
        for (int kc = 0; kc < 4; ++kc) {
            const int kk = 4 * kc + 2 * half;
            v2f pa = (v2f){ pbuf[l15 * 17 + kk], pbuf[l15 * 17 + kk + 1] };   // A: M=l15
            const float* v0 = vb + (size_t)(kt * 16 + kk) * ROWSTRIDE;        // B K rows
            #pragma unroll
            for (int dt = 0; dt < 4; ++dt) {
                v2f bfrag = (v2f){ v0[dt * 16 + l15],
                                   v0[ROWSTRIDE + dt * 16 + l15] };           // N = d col
                ctxacc[dt] = __builtin_amdgcn_wmma_f32_16x16x4_f32(false, pa, false, bfrag,
                                                                   (short)0, ctxacc[dt],
                                                                   false, false);
            }
        }
        __syncthreads();   // pbuf reused next tile
    }

    // ---- write context tile (coalesced 64B per half-wave row) ----
    #pragma unroll
    for (int dt = 0; dt < 4; ++dt) {
        #pragma unroll
        for (int r = 0; r < 8; ++r) {
            cb[(size_t)(r + 8 * half) * ROWSTRIDE + dt * 16 + l15] = ctxacc[dt][r];
        }
    }
}

extern "C" void kernel_launch(void* const* d_in, const int* in_sizes, int n_in,
                              void* d_out, int out_size, void* d_ws, size_t ws_size,
                              hipStream_t stream) {
    const float* q = (const float*)d_in[0];
    const float* k = (const float*)d_in[1];
    const float* v = (const float*)d_in[2];
    float* out  = (float*)d_out;
    float* ctx  = out;                                  // [2,2048,16,64]
    float* attn = out + (size_t)2 * 2048 * 16 * 64;     // [2,16,2048,2048]

    dim3 grid(2 * HEADS * KTILES);   // 4096 single-wave workgroups
    attn_fwd_wmma<<<grid, 32, 0, stream>>>(q, k, v, ctx, attn);
}